// BrainTextModel_5875515261528
// MI455X (gfx1250) — compile-verified
//
#include <hip/hip_runtime.h>
#include <hip/hip_bf16.h>
#include <stdint.h>

// Problem dims (match reference)
#define B_ 32
#define T_ 256
#define E_ 512
#define H_ 1024
#define V_ 50257

typedef __attribute__((ext_vector_type(2))) float v2f;
typedef __attribute__((ext_vector_type(8))) float v8f;
typedef __attribute__((ext_vector_type(4))) int   v4i;
typedef __attribute__((ext_vector_type(2))) int   v2i;

// ---- CDNA5 async global->LDS path (guarded; falls back to plain loads) ----
#if __has_builtin(__builtin_amdgcn_global_load_async_to_lds_b64) && \
    __has_builtin(__builtin_amdgcn_global_load_async_to_lds_b128)
#define ASYNC_LDS_OK 1
// Builtins take typed AS-qualified vector pointers:
//   b128: (AS1 v4i* src, AS3 v4i* dst, imm offset, imm cpol)
//   b64 : (AS1 v2i* src, AS3 v2i* dst, imm offset, imm cpol)
typedef __attribute__((address_space(1))) v4i g_v4i;
typedef __attribute__((address_space(3))) v4i l_v4i;
typedef __attribute__((address_space(1))) v2i g_v2i;
typedef __attribute__((address_space(3))) v2i l_v2i;
#if __has_builtin(__builtin_amdgcn_s_wait_asynccnt)
#define WAIT_ASYNC() __builtin_amdgcn_s_wait_asynccnt(0)
#else
#define WAIT_ASYNC() asm volatile("s_wait_asynccnt 0" ::: "memory")
#endif
#else
#define ASYNC_LDS_OK 0
#endif

// fp32 WMMA: D(16x16,f32) = A(16x4,f32) x B(4x16,f32) + C
// 8-arg form: (neg_a, A, neg_b, B, c_mod, C, reuse_a, reuse_b)
__device__ __forceinline__ v8f wmma_f32_k4(v2f a, v2f b, v8f c) {
    return __builtin_amdgcn_wmma_f32_16x16x4_f32(false, a, false, b,
                                                 (short)0, c, false, false);
}

// =====================================================================
// Kernel 1: fused  embed-gather -> fc GEMM (WMMA f32) -> LIF scan
// grid = (H/32, B); block = 256 threads (8 waves)
// Block (hx, b): rows = all T=256 steps of batch b, cols = 32 hidden units.
// =====================================================================
#define SA 20   // LDS row stride (floats) for A/B tiles: 20 -> conflict-free frag reads
#define SC 34   // LDS row stride for currents tile (32 cols + pad)

__global__ __launch_bounds__(256) void snn_embed_fc_lif(
    const int* __restrict__ tokens, const float* __restrict__ emb,
    const float* __restrict__ fc_w, const float* __restrict__ fc_b,
    float* __restrict__ ws_mem)
{
    __shared__ union {
        struct { float A[T_ * SA]; float Bt[32 * SA]; } g;  // GEMM staging
        float cur[T_ * SC];                                 // currents tile
    } sm;

    const int tid  = threadIdx.x;
    const int wave = tid >> 5;
    const int lane = tid & 31;
    const int b    = blockIdx.y;
    const int h0   = blockIdx.x * 32;

    // Each thread owns gather-row t == tid (one timestep's embedding row)
    const int tok = tokens[b * T_ + tid];
    const float* arow = emb + (size_t)tok * E_;

    // fc_w staging assignment: 32 rows x 8 k-pairs
    const int bn = tid >> 3;   // 0..31: hidden unit within tile
    const int bq = tid & 7;    // 0..7 : k-pair
    const float* brow = fc_w + (size_t)(h0 + bn) * E_;

    const v8f vzero = {0.f,0.f,0.f,0.f,0.f,0.f,0.f,0.f};
    v8f acc[2][2];
    acc[0][0] = vzero; acc[0][1] = vzero; acc[1][0] = vzero; acc[1][1] = vzero;

    // Fragment addressing (32-bit WMMA layouts):
    //  A: lane l holds A[l%16][2*(l/16)+j] in vgpr j
    //  B: lane l holds B[2*(l/16)+j][l%16] in vgpr j
    const int mrow0 = (2 * wave) * 16 + (lane & 15);
    const int mrow1 = mrow0 + 16;
    const int khalf = 2 * (lane >> 4);
    const int ncol0 = (lane & 15);
    const int ncol1 = 16 + (lane & 15);

    for (int kc = 0; kc < E_; kc += 16) {
#if ASYNC_LDS_OK
        // Gathered embedding rows: per-lane async DMA directly into LDS.
        #pragma unroll
        for (int q = 0; q < 4; ++q)
            __builtin_amdgcn_global_load_async_to_lds_b128(
                (g_v4i*)(arow + kc + q * 4),
                (l_v4i*)&sm.g.A[tid * SA + q * 4], 0, 0);
        __builtin_amdgcn_global_load_async_to_lds_b64(
            (g_v2i*)(brow + kc + bq * 2),
            (l_v2i*)&sm.g.Bt[bn * SA + bq * 2], 0, 0);
        WAIT_ASYNC();
#else
        #pragma unroll
        for (int q = 0; q < 4; ++q) {
            float4 va = *(const float4*)(arow + kc + q * 4);
            *(float4*)&sm.g.A[tid * SA + q * 4] = va;   // (tid*20+q*4)*4B: 16B aligned
        }
        v2f vb = *(const v2f*)(brow + kc + bq * 2);
        *(v2f*)&sm.g.Bt[bn * SA + bq * 2] = vb;
#endif
        __syncthreads();

        #pragma unroll
        for (int kk = 0; kk < 16; kk += 4) {
            const int ko = kk + khalf;
            v2f a0 = *(const v2f*)&sm.g.A[mrow0 * SA + ko];
            v2f a1 = *(const v2f*)&sm.g.A[mrow1 * SA + ko];
            v2f b0 = *(const v2f*)&sm.g.Bt[ncol0 * SA + ko];
            v2f b1 = *(const v2f*)&sm.g.Bt[ncol1 * SA + ko];
            acc[0][0] = wmma_f32_k4(a0, b0, acc[0][0]);
            acc[0][1] = wmma_f32_k4(a0, b1, acc[0][1]);
            acc[1][0] = wmma_f32_k4(a1, b0, acc[1][0]);
            acc[1][1] = wmma_f32_k4(a1, b1, acc[1][1]);
        }
        __syncthreads();
    }

    // Spill currents tile to LDS (C/D layout: vgpr r, lane l -> D[r+8*(l/16)][l%16])
    #pragma unroll
    for (int mi = 0; mi < 2; ++mi) {
        const int rbase = (2 * wave + mi) * 16 + 8 * (lane >> 4);
        #pragma unroll
        for (int ni = 0; ni < 2; ++ni) {
            const int col = ni * 16 + (lane & 15);
            #pragma unroll
            for (int r = 0; r < 8; ++r)
                sm.cur[(rbase + r) * SC + col] = acc[mi][ni][r];
        }
    }
    __syncthreads();

    // LIF scan over t (sequential), parallel over the 32 hidden units.
    // Only final membrane potential is needed (spikes are discarded upstream).
    if (tid < 32) {
        const int hg = h0 + tid;
        const float bias = fc_b[hg];
        float m = 0.0f;  // RESET
        for (int t = 0; t < T_; ++t) {
            const float c = sm.cur[t * SC + tid] + bias;
            m = 0.9f * m + 0.1f * c;          // beta*mem + (1-beta)*cur
            m = (m >= 1.0f) ? 0.0f : m;       // fire + reset-to-zero
        }
        ws_mem[b * H_ + hg] = m;
    }
}

// =====================================================================
// Kernel 2: logits = final_mem @ out_w^T + out_b   (memory-bound: 206MB stream)
// One 16-vocab tile per wave; 2 WMMA accumulators cover all 32 batch rows.
// =====================================================================
__global__ __launch_bounds__(256) void snn_readout(
    const float* __restrict__ mem, const float* __restrict__ out_w,
    const float* __restrict__ out_b, float* __restrict__ logits)
{
    const int tid  = threadIdx.x;
    const int wave = tid >> 5;
    const int lane = tid & 31;
    const int tile = blockIdx.x * 8 + wave;
    const int v0   = tile * 16;
    if (v0 >= V_) return;

    const int vr     = v0 + (lane & 15);
    const int vclamp = (vr < V_) ? vr : (V_ - 1);
    const float* wrow = out_w + (size_t)vclamp * H_;   // B-frag: out_w[v][k], contiguous in k
    const float* m0   = mem + (size_t)(lane & 15) * H_; // A-frag rows b=0..15
    const float* m1   = m0 + 16 * H_;                   // b=16..31
    const int lk = 2 * (lane >> 4);

    const v8f vzero = {0.f,0.f,0.f,0.f,0.f,0.f,0.f,0.f};
    v8f acc0 = vzero, acc1 = vzero;

    for (int k0 = 0; k0 < H_; k0 += 64) {
        __builtin_prefetch(wrow + k0 + 256, 0, 0);   // stream out_w ~1KB ahead
        #pragma unroll
        for (int kk = 0; kk < 64; kk += 4) {
            const int kA = k0 + kk + lk;
            v2f bw = *(const v2f*)(wrow + kA);
            v2f a0 = *(const v2f*)(m0 + kA);
            v2f a1 = *(const v2f*)(m1 + kA);
            acc0 = wmma_f32_k4(a0, bw, acc0);
            acc1 = wmma_f32_k4(a1, bw, acc1);
        }
    }

    if (vr < V_) {
        const float bias = out_b[vr];
        const int bb = 8 * (lane >> 4);
        #pragma unroll
        for (int r = 0; r < 8; ++r) {
            logits[(size_t)(bb + r) * V_ + vr]      = acc0[r] + bias;
            logits[(size_t)(bb + r + 16) * V_ + vr] = acc1[r] + bias;
        }
    }
}

// =====================================================================
extern "C" void kernel_launch(void* const* d_in, const int* in_sizes, int n_in,
                              void* d_out, int out_size, void* d_ws, size_t ws_size,
                              hipStream_t stream) {
    (void)in_sizes; (void)n_in; (void)out_size; (void)ws_size;
    const int*   tokens = (const int*)d_in[0];
    const float* emb    = (const float*)d_in[1];
    const float* fc_w   = (const float*)d_in[2];
    const float* fc_b   = (const float*)d_in[3];
    const float* out_w  = (const float*)d_in[4];
    const float* out_b  = (const float*)d_in[5];
    float* lg     = (float*)d_out;
    float* ws_mem = (float*)d_ws;   // 32*1024 floats: final membrane potentials

    dim3 g1(H_ / 32, B_);           // (32, 32) blocks
    snn_embed_fc_lif<<<g1, 256, 0, stream>>>(tokens, emb, fc_w, fc_b, ws_mem);

    const int ntiles = (V_ + 15) / 16;     // 3142
    const int nblk   = (ntiles + 7) / 8;   // 393
    snn_readout<<<nblk, 256, 0, stream>>>(ws_mem, out_w, out_b, lg);
}